// SplitLinear_34050500723019
// MI455X (gfx1250) — compile-verified
//
#include <hip/hip_runtime.h>
#include <hip/hip_bf16.h>

// CDNA5 / gfx1250, wave32. Uses V_WMMA_F32_16X16X4_F32.
//
// Problem: out[b, o*2500 + c] = sum_s x[b, s*2500 + c] * w[o, s, c] + bias
//   B=64, O=4, C=2500, S=400.  Memory bound: x = 256MB streamed once.
//
// WMMA mapping (per wave, per K-step s):
//   A[m,k] = x[b0+m, s*2500 + c0 + k]           (16x4, k = chunk-in-group)
//   B[k,n] = (n&3 == k) ? w[n>>2, s, c0+k] : 0  (4x16, block diagonal)
//   D[m,n] accumulates out[b0+m, o=n>>2, c=c0+(n&3)]
//
// Fragment layouts per ISA 7.12.2 (32-bit, wave32):
//   A 16x4 : lane L holds row m=L&15; VGPR0=K=(L<16?0:2), VGPR1=K=(L<16?1:3)
//   B 4x16 : lane L holds col n=L&15; VGPR0=K=(L<16?0:2), VGPR1=K=(L<16?1:3)
//   D 16x16: lane L holds col n=L&15; VGPR r = row m = r + (L<16?0:8)

typedef __attribute__((ext_vector_type(2))) float v2f;
typedef __attribute__((ext_vector_type(8))) float v8f;

#define IN_FEATURES 1000000
#define OUT_SETS    4
#define NUM_CHUNKS  2500
#define SPLIT       400
#define BATCH       64
#define OUT_COLS    (OUT_SETS * NUM_CHUNKS)   // 10000

__global__ __launch_bounds__(256)
void split_linear_wmma_f32(const float* __restrict__ x,
                           const float* __restrict__ w,
                           const float* __restrict__ bias,
                           float* __restrict__ out) {
    const int tid  = threadIdx.x;
    const int wave = tid >> 5;        // 8 waves per block
    const int lane = tid & 31;

    // Each wave: one 4-chunk group. Block covers 32 chunks.
    const int c0 = blockIdx.x * 32 + wave * 4;
    if (c0 >= NUM_CHUNKS) return;     // wave-uniform: EXEC stays all-ones

    const int b0   = blockIdx.y * 16; // 16-row batch tile
    const int m    = lane & 15;       // A row / B,D column index
    const int half = lane >> 4;       // 0: K={0,1}, 1: K={2,3}
    const int o    = m >> 2;          // output set for this lane's column
    const int km   = m & 3;           // chunk-in-group for this lane's column

    // A: lane loads x[b0+m, s*2500 + c0 + half*2 .. +1] as one b64
    const float* xp = x + (size_t)(b0 + m) * IN_FEATURES + (size_t)c0 + (size_t)(half * 2);
    // B: lane's (possible) weight w[o, s, c0+km]
    const float* wp = w + (size_t)o * (SPLIT * NUM_CHUNKS) + (size_t)c0 + (size_t)km;

    // Which K-slot (if any) of this lane's B fragment is on the diagonal:
    const bool useX = (km == half * 2);
    const bool useY = (km == half * 2 + 1);

    v8f acc0 = {0.f, 0.f, 0.f, 0.f, 0.f, 0.f, 0.f, 0.f};
    v8f acc1 = {0.f, 0.f, 0.f, 0.f, 0.f, 0.f, 0.f, 0.f};

    #pragma unroll 4
    for (int s = 0; s < SPLIT; s += 2) {
        // --- step s ---
        v2f a0 = *(const v2f*)(xp + (size_t)s * NUM_CHUNKS);
        float wv0 = wp[(size_t)s * NUM_CHUNKS];
        v2f b0v;
        b0v.x = useX ? wv0 : 0.0f;
        b0v.y = useY ? wv0 : 0.0f;
        __builtin_prefetch(xp + (size_t)(s + 16) * NUM_CHUNKS, 0, 3);
        acc0 = __builtin_amdgcn_wmma_f32_16x16x4_f32(
            false, a0, false, b0v, (short)0, acc0, false, false);

        // --- step s+1 (independent accumulator -> overlap WMMA latency) ---
        v2f a1 = *(const v2f*)(xp + (size_t)(s + 1) * NUM_CHUNKS);
        float wv1 = wp[(size_t)(s + 1) * NUM_CHUNKS];
        v2f b1v;
        b1v.x = useX ? wv1 : 0.0f;
        b1v.y = useY ? wv1 : 0.0f;
        acc1 = __builtin_amdgcn_wmma_f32_16x16x4_f32(
            false, a1, false, b1v, (short)0, acc1, false, false);
    }

    const float bv = bias[o * NUM_CHUNKS + c0 + km];
    const size_t colOff = (size_t)o * NUM_CHUNKS + (size_t)c0 + (size_t)km;

    #pragma unroll
    for (int r = 0; r < 8; ++r) {
        const int brow = b0 + half * 8 + r;   // D row for VGPR r in this half-wave
        out[(size_t)brow * OUT_COLS + colOff] = acc0[r] + acc1[r] + bv;
    }
}

extern "C" void kernel_launch(void* const* d_in, const int* in_sizes, int n_in,
                              void* d_out, int out_size, void* d_ws, size_t ws_size,
                              hipStream_t stream) {
    (void)in_sizes; (void)n_in; (void)d_ws; (void)ws_size; (void)out_size;
    const float* x    = (const float*)d_in[0];   // [64, 1000000]
    const float* wgt  = (const float*)d_in[1];   // [4, 400, 2500]
    const float* bias = (const float*)d_in[2];   // [10000]
    float* out = (float*)d_out;                  // [64, 10000]

    // 625 chunk-groups of 4 chunks; 8 groups (32 chunks) per block -> 79 blocks
    // (last block's surplus waves self-skip). y: 4 batch tiles of 16.
    dim3 grid(79, 4, 1);
    dim3 block(256, 1, 1);
    split_linear_wmma_f32<<<grid, block, 0, stream>>>(x, wgt, bias, out);
}